// L2Quantizer_39092792328257
// MI455X (gfx1250) — compile-verified
//
#include <hip/hip_runtime.h>
#include <hip/hip_bf16.h>

typedef float v2f __attribute__((ext_vector_type(2)));
typedef float v8f __attribute__((ext_vector_type(8)));

#define BB    8
#define DIN   192
#define HW    1024      // 32*32
#define KK    4096
#define DH    32
#define NPIX  8192      // B*H*W

// ---------------------------------------------------------------------------
// Kernel 1: raw[n][o] = sum_i latent[b][i][hw] * wv_w[o][i] + wv_b[o]
//           x2[n]     = ||raw[n]||^2
// One wave per pixel (lane = output channel o), 8 pixels per 256-thread block.
// ---------------------------------------------------------------------------
__global__ __launch_bounds__(256) void k_raw(const float* __restrict__ latent,
                                             const float* __restrict__ wv_w,
                                             const float* __restrict__ wv_b,
                                             float* __restrict__ rawOut,   // [NPIX][DH]
                                             float* __restrict__ x2)       // [NPIX]
{
    __shared__ float wvs[DH * 193];   // padded: stride 193 -> conflict-free
    __shared__ float qs[DIN * 8];     // [i][p]
    const int t = threadIdx.x;

    for (int idx = t; idx < DH * DIN; idx += 256) {
        int o = idx / DIN, i = idx - o * DIN;
        wvs[o * 193 + i] = wv_w[idx];
    }
    const int nBase  = blockIdx.x * 8;
    const int b      = nBase / HW;
    const int hwBase = nBase - b * HW;
    for (int idx = t; idx < DIN * 8; idx += 256) {
        int i = idx >> 3, p = idx & 7;
        qs[i * 8 + p] = latent[(b * DIN + i) * HW + hwBase + p];
    }
    __syncthreads();

    const int wave = t >> 5;
    const int lane = t & 31;
    const int n    = nBase + wave;

    float acc = wv_b[lane];
    #pragma unroll 4
    for (int i = 0; i < DIN; ++i)
        acc += qs[i * 8 + wave] * wvs[lane * 193 + i];   // qs read is broadcast

    rawOut[n * DH + lane] = acc;

    float sq = acc * acc;
    #pragma unroll
    for (int off = 16; off > 0; off >>= 1)
        sq += __shfl_xor(sq, off, 32);
    if (lane == 0) x2[n] = sq;
}

// ---------------------------------------------------------------------------
// Kernel 2: c2[k] = ||codebook[k]||^2, plus passthrough copy of codebook.
// ---------------------------------------------------------------------------
__global__ __launch_bounds__(256) void k_c2(const float* __restrict__ codebook,
                                            float* __restrict__ c2,
                                            float* __restrict__ cbOut)
{
    const int k = blockIdx.x * 256 + threadIdx.x;
    if (k >= KK) return;
    float s = 0.f;
    #pragma unroll
    for (int d = 0; d < DH; ++d) {
        float v = codebook[k * DH + d];
        cbOut[k * DH + d] = v;
        s += v * v;
    }
    c2[k] = s;
}

// ---------------------------------------------------------------------------
// Kernel 3: fused logits + dual argmax + quantized gather.
//   Block = 16-pixel tile, 8 waves; wave handles a 16-wide K column chunk,
//   32 outer iterations cover K=4096.  inter = raw(16x32) x codebookT(32x16)
//   via 8 chained V_WMMA_F32_16X16X4_F32.
// ---------------------------------------------------------------------------
__global__ __launch_bounds__(256) void k_logits(
    const float* __restrict__ rawIn,      // [NPIX][DH]
    const float* __restrict__ codebook,   // [KK][DH]
    const float* __restrict__ gumbel,     // [NPIX][KK]
    const float* __restrict__ x2,         // [NPIX]
    const float* __restrict__ c2,         // [KK]
    const float* __restrict__ temp1,      // scalar
    const int*   __restrict__ temperature,// scalar
    float* __restrict__ logitOut,         // [NPIX][KK]
    int*   __restrict__ codeOut,          // [NPIX]
    int*   __restrict__ trueOut,          // [NPIX]
    float* __restrict__ quantOut)         // [NPIX][DH]
{
    __shared__ float As[16 * 33];         // padded raw tile
    __shared__ float x2s[16];
    __shared__ float redV[8][16];  __shared__ int redI[8][16];
    __shared__ float redGV[8][16]; __shared__ int redGI[8][16];

    const int t       = threadIdx.x;
    const int wave    = t >> 5;
    const int lane    = t & 31;
    const int pixBase = blockIdx.x * 16;

    for (int idx = t; idx < 16 * DH; idx += 256) {
        int m = idx >> 5, d = idx & 31;
        As[m * 33 + d] = rawIn[(pixBase + m) * DH + d];
    }
    if (t < 16) x2s[t] = x2[pixBase + t];
    __syncthreads();

    const float s    = 1.0f / (temp1[0] * (float)temperature[0]);
    const int   half = lane >> 4;           // 0: rows m=r, 1: rows m=r+8
    const int   mOff = half * 8;
    const int   nLoc = lane & 15;           // column within 16-wide chunk
    const int   koff = half * 2;            // WMMA K sub-slot
    const int   mA   = lane & 15;           // A-fragment row

    // Preload A fragments (8 K-steps x 2 values) and x2 per output row.
    float a0[8], a1[8], x2r[8];
    #pragma unroll
    for (int stp = 0; stp < 8; ++stp) {
        a0[stp] = As[mA * 33 + 4 * stp + koff];
        a1[stp] = As[mA * 33 + 4 * stp + koff + 1];
    }
    #pragma unroll
    for (int r = 0; r < 8; ++r) x2r[r] = x2s[mOff + r];

    float bestV[8], bestGV[8]; int bestI[8], bestGI[8];
    #pragma unroll
    for (int r = 0; r < 8; ++r) {
        bestV[r] = -3.4e38f; bestGV[r] = -3.4e38f;
        bestI[r] = 0;        bestGI[r] = 0;
    }

    for (int it = 0; it < 32; ++it) {
        const int   kg   = it * 128 + wave * 16 + nLoc;  // this lane's column
        const float c2v  = c2[kg];
        const float* crow = codebook + (size_t)kg * DH;

        v8f acc = {};
        #pragma unroll
        for (int stp = 0; stp < 8; ++stp) {
            v2f a, bf;
            a.x  = a0[stp];
            a.y  = a1[stp];
            bf.x = crow[4 * stp + koff];
            bf.y = crow[4 * stp + koff + 1];
            acc = __builtin_amdgcn_wmma_f32_16x16x4_f32(
                false, a, false, bf, (short)0, acc, false, false);
        }

        #pragma unroll
        for (int r = 0; r < 8; ++r) {
            const size_t npx = (size_t)(pixBase + mOff + r);
            const float  l   = 2.0f * acc[r] - x2r[r] - c2v;   // logitRaw
            logitOut[npx * KK + kg] = l;
            const float g  = gumbel[npx * KK + kg];
            const float gv = l * s + g;                        // scaled+gumbel
            if (l  > bestV[r])  { bestV[r]  = l;  bestI[r]  = kg; }
            if (gv > bestGV[r]) { bestGV[r] = gv; bestGI[r] = kg; }
        }
    }

    // Reduce across the 16 lanes sharing each output row (xor<16 stays in half).
    #pragma unroll
    for (int r = 0; r < 8; ++r) {
        #pragma unroll
        for (int off = 8; off > 0; off >>= 1) {
            float ov  = __shfl_xor(bestV[r],  off, 32);
            int   oi  = __shfl_xor(bestI[r],  off, 32);
            if (ov > bestV[r] || (ov == bestV[r] && oi < bestI[r])) {
                bestV[r] = ov; bestI[r] = oi;
            }
            float ogv = __shfl_xor(bestGV[r], off, 32);
            int   ogi = __shfl_xor(bestGI[r], off, 32);
            if (ogv > bestGV[r] || (ogv == bestGV[r] && ogi < bestGI[r])) {
                bestGV[r] = ogv; bestGI[r] = ogi;
            }
        }
    }
    if (nLoc == 0) {
        #pragma unroll
        for (int r = 0; r < 8; ++r) {
            redV [wave][mOff + r] = bestV[r];  redI [wave][mOff + r] = bestI[r];
            redGV[wave][mOff + r] = bestGV[r]; redGI[wave][mOff + r] = bestGI[r];
        }
    }
    __syncthreads();

    if (t < 16) {
        float bv = -3.4e38f, bgv = -3.4e38f; int bi = 0, bgi = 0;
        for (int w = 0; w < 8; ++w) {
            float v = redV[w][t];  int i  = redI[w][t];
            if (v > bv || (v == bv && i < bi)) { bv = v; bi = i; }
            float gv = redGV[w][t]; int gi = redGI[w][t];
            if (gv > bgv || (gv == bgv && gi < bgi)) { bgv = gv; bgi = gi; }
        }
        const int npx = pixBase + t;
        trueOut[npx] = bi;
        codeOut[npx] = bgi;
        // quantized = sample @ codebook = codebook[code]  (sample is one-hot)
        #pragma unroll
        for (int d = 0; d < DH; ++d)
            quantOut[npx * DH + d] = codebook[bgi * DH + d];
    }
}

// ---------------------------------------------------------------------------
// Kernel 4: hard[b][o][hw] = sum_d quantized[n][d] * wq_w[o][d] + wq_b[o]
// 64 pixels per block; wq_w + quantized staged in LDS; coalesced NCHW writes.
// ---------------------------------------------------------------------------
__global__ __launch_bounds__(256) void k_hard(const float* __restrict__ quant,  // [NPIX][DH]
                                              const float* __restrict__ wq_w,   // [DIN][DH]
                                              const float* __restrict__ wq_b,   // [DIN]
                                              float* __restrict__ hard)         // [B][DIN][HW]
{
    __shared__ float wqs[DIN * DH];   // broadcast reads: no padding needed
    __shared__ float qs[64 * 33];     // padded
    __shared__ float bs[DIN];
    const int t = threadIdx.x;

    for (int idx = t; idx < DIN * DH; idx += 256) wqs[idx] = wq_w[idx];
    for (int idx = t; idx < DIN; idx += 256)      bs[idx]  = wq_b[idx];

    const int nBase  = blockIdx.x * 64;
    for (int idx = t; idx < 64 * DH; idx += 256) {
        int p = idx >> 5, d = idx & 31;
        qs[p * 33 + d] = quant[(nBase + p) * DH + d];
    }
    __syncthreads();

    const int b      = nBase / HW;
    const int hwBase = nBase - b * HW;
    const int p      = t & 63;       // pixel within tile
    const int oq     = t >> 6;       // output-channel phase (0..3)

    for (int j = 0; j < 48; ++j) {
        const int o = j * 4 + oq;    // uniform per wave -> wqs reads broadcast
        float acc = bs[o];
        #pragma unroll
        for (int d = 0; d < DH; ++d)
            acc += qs[p * 33 + d] * wqs[o * DH + d];
        hard[(size_t)(b * DIN + o) * HW + hwBase + p] = acc;
    }
}

// ---------------------------------------------------------------------------
extern "C" void kernel_launch(void* const* d_in, const int* in_sizes, int n_in,
                              void* d_out, int out_size, void* d_ws, size_t ws_size,
                              hipStream_t stream) {
    const float* latent      = (const float*)d_in[0];
    const float* codebook    = (const float*)d_in[1];
    const float* wv_w        = (const float*)d_in[2];
    const float* wv_b        = (const float*)d_in[3];
    const float* wq_w        = (const float*)d_in[4];
    const float* wq_b        = (const float*)d_in[5];
    const float* temp1       = (const float*)d_in[6];
    const float* gumbel      = (const float*)d_in[7];
    const int*   temperature = (const int*)d_in[8];

    // Outputs concatenated flat in return order (all 4-byte elements):
    float* out      = (float*)d_out;
    float* hardOut  = out;                         // 8*192*32*32 = 1,572,864
    int*   codeOut  = (int*)(out + 1572864);       // 8,192
    int*   trueOut  = (int*)(out + 1581056);       // 8,192
    float* logitOut = out + 1589248;               // 8192*4096 = 33,554,432
    float* rawOut   = out + 35143680;              // 262,144
    float* quantOut = out + 35405824;              // 262,144
    float* cbOut    = out + 35667968;              // 131,072

    float* x2 = (float*)d_ws;                      // [NPIX]
    float* c2 = x2 + NPIX;                         // [KK]

    k_raw   <<<NPIX / 8, 256, 0, stream>>>(latent, wv_w, wv_b, rawOut, x2);
    k_c2    <<<KK / 256, 256, 0, stream>>>(codebook, c2, cbOut);
    k_logits<<<NPIX / 16, 256, 0, stream>>>(rawOut, codebook, gumbel, x2, c2,
                                            temp1, temperature,
                                            logitOut, codeOut, trueOut, quantOut);
    k_hard  <<<NPIX / 64, 256, 0, stream>>>(quantOut, wq_w, wq_b, hardOut);
}